// CrossNet_67551245631681
// MI455X (gfx1250) — compile-verified
//
#include <hip/hip_runtime.h>
#include <hip/hip_bf16.h>

// ---------------------------------------------------------------------------
// CrossNet on MI455X (gfx1250).
//
// Algebra: xl = A_l * x0 + B_l  (A_l scalar/row, B_l = partial bias sum,
// identical for every row). The network reduces to:
//   d[l]  = x0 . W[l]                (per row; [B,1024]x[1024,4] fp32 GEMM)
//   c[l]  = (sum_{j<l} b[j]) . W[l]  (4 scalars)
//   A=1; for l: s = A*d[l] + c[l]; A += s;
//   out   = A * x0 + (b0+b1+b2+b3)
//
// The GEMM runs on V_WMMA_F32_16X16X4_F32 (fp32 throughout). HBM traffic is
// ~128 MB => ~5.5 us floor at 23.3 TB/s; x0 (64 MB) fits in the 192 MB L2 so
// the second pass reads mostly from L2.
//
// K2 inner loop (per WMMA step): 1x global_load_b64 (A fragment, contiguous
// float2 per lane per the 16x4 fp32 A layout), 1x ds_load_b64 from a
// zero-padded B-fragment table built once per block in LDS (no EXEC churn,
// no re-fetch of W), 1x v_wmma. Two accumulator chains for ILP.
// ---------------------------------------------------------------------------

typedef __attribute__((ext_vector_type(2))) float v2f;
typedef __attribute__((ext_vector_type(4))) float v4f;
typedef __attribute__((ext_vector_type(8))) float v8f;

#define BATCH    16384
#define DIM      1024
#define NLAYERS  4
#define KCHUNKS  4                      // split K=1024 for occupancy
#define KCHUNK   (DIM / KCHUNKS)        // 256
#define KSTEPS   (KCHUNK / 4)           // 64 WMMA steps per wave
#define ROWTILES (BATCH / 16)           // 1024 WMMA row tiles
#define WPB      8                      // waves per block

// workspace layout (float offsets); total = 32768 + 4*16384*4 floats ~ 1.2 MB
#define WS_C     0                      // c[4]
#define WS_BSUM  16                     // Bsum[1024]
#define WS_A4    2048                   // A4[16384]
#define WS_DPART 32768                  // dpart[KCHUNKS][BATCH][4]

// --- K1: c[l] and Bsum (1 block, 1024 threads) -----------------------------
__global__ void crossnet_prep(const float* __restrict__ W,
                              const float* __restrict__ b,
                              float* __restrict__ ws) {
  const int t = threadIdx.x;            // 0..1023
  const float b0 = b[0 * DIM + t], b1 = b[1 * DIM + t];
  const float b2 = b[2 * DIM + t], b3 = b[3 * DIM + t];
  const float B1 = b0, B2 = b0 + b1, B3 = b0 + b1 + b2;
  ws[WS_BSUM + t] = B3 + b3;            // Bsum

  __shared__ float red[3][1024];
  red[0][t] = B1 * W[1 * DIM + t];
  red[1][t] = B2 * W[2 * DIM + t];
  red[2][t] = B3 * W[3 * DIM + t];
  __syncthreads();
  for (int off = 512; off > 0; off >>= 1) {
    if (t < off) {
      red[0][t] += red[0][t + off];
      red[1][t] += red[1][t + off];
      red[2][t] += red[2][t + off];
    }
    __syncthreads();
  }
  if (t == 0) {
    ws[WS_C + 0] = 0.0f;
    ws[WS_C + 1] = red[0][0];
    ws[WS_C + 2] = red[1][0];
    ws[WS_C + 3] = red[2][0];
  }
}

// --- K2: dpart[kchunk][row][l] via v_wmma_f32_16x16x4_f32 ------------------
// Block = 8 waves, all on the same K chunk (shared B table); wave w handles
// row tile (blockIdx>>2)*8 + w. A-fragment per the WMMA ISA layout: lane L
// holds row (L&15), K pair offset 2*(L>>4) -> one contiguous float2 load.
__global__ void __launch_bounds__(256)
crossnet_dots(const float* __restrict__ x0,
              const float* __restrict__ W,
              float* __restrict__ ws) {
  const int kchunk  = blockIdx.x & (KCHUNKS - 1);
  const int waveId  = threadIdx.x >> 5;
  const int rowTile = (blockIdx.x >> 2) * WPB + waveId;
  const int lane    = threadIdx.x & 31;
  const int n       = lane & 15;        // row (A) / column (B)
  const int hi      = lane >> 4;        // K half: +0 or +2

  // Zero-padded B-fragment table: btab[step][lane] = float2 B fragment.
  // Lanes n<4 carry W[n][kchunk*256 + step*4 + 2*hi .. +1]; others are 0.
  __shared__ v2f btab[KSTEPS][32];      // 16 KB

  {
    const int t = threadIdx.x;
#pragma unroll
    for (int i = 0; i < (KSTEPS * 32) / 256; ++i) {   // 8 entries/thread
      const int e     = t + i * 256;
      const int estep = e >> 5;
      const int elane = e & 31;
      const int en    = elane & 15;
      const int ekk   = (elane >> 4) * 2;
      const int kpos  = kchunk * KCHUNK + estep * 4 + ekk;
      float vx = W[(en & 3) * DIM + kpos];      // clamped, branchless
      float vy = W[(en & 3) * DIM + kpos + 1];
      if (en >= 4) { vx = 0.0f; vy = 0.0f; }    // v_cndmask, no load branch
      v2f v; v.x = vx; v.y = vy;
      btab[estep][elane] = v;
    }
  }
  __syncthreads();

  const float* aptr = x0 + (size_t)(rowTile * 16 + n) * DIM
                         + kchunk * KCHUNK + hi * 2;
  const v2f* bp = &btab[0][lane];       // advance by 32 entries per step

  v8f acc0 = {}, acc1 = {};
#pragma unroll 4
  for (int s = 0; s < KSTEPS; s += 2) {
    v2f a0 = *(const v2f*)(aptr + s * 4);
    v2f a1 = *(const v2f*)(aptr + s * 4 + 4);
    v2f b0 = bp[(s + 0) * 32];
    v2f b1 = bp[(s + 1) * 32];
    acc0 = __builtin_amdgcn_wmma_f32_16x16x4_f32(
        false, a0, false, b0, (short)0, acc0, false, false);
    acc1 = __builtin_amdgcn_wmma_f32_16x16x4_f32(
        false, a1, false, b1, (short)0, acc1, false, false);
  }
  v8f acc = acc0 + acc1;

  // C layout: VGPR r -> row rowTile*16 + hi*8 + r, column n (n<4 useful)
  if (n < 4) {
    float* dp = ws + WS_DPART + (size_t)kchunk * BATCH * 4;
#pragma unroll
    for (int r = 0; r < 8; ++r) {
      const int row = rowTile * 16 + hi * 8 + r;
      dp[(size_t)row * 4 + n] = acc[r];
    }
  }
}

// --- K3: fold K chunks, run the 4-step scalar recurrence -> A4[row] --------
__global__ void crossnet_rec(float* __restrict__ ws) {
  const int row = blockIdx.x * blockDim.x + threadIdx.x;
  if (row >= BATCH) return;
  v4f d = {0.0f, 0.0f, 0.0f, 0.0f};
  const float* dp = ws + WS_DPART;
#pragma unroll
  for (int cch = 0; cch < KCHUNKS; ++cch)
    d += *(const v4f*)(dp + ((size_t)cch * BATCH + row) * 4);

  float A = 1.0f;
#pragma unroll
  for (int l = 0; l < NLAYERS; ++l) {
    const float s = A * d[l] + ws[WS_C + l];
    A += s;
  }
  ws[WS_A4 + row] = A;
}

// --- K4: out = A4[row] * x0 + Bsum  (coalesced float4, x0 from L2) ---------
__global__ void crossnet_out(const float* __restrict__ x0,
                             const float* __restrict__ ws,
                             float* __restrict__ out) {
  const size_t i4  = (size_t)blockIdx.x * blockDim.x + threadIdx.x;
  const size_t j   = i4 * 4;
  const int    row = (int)(j >> 10);    // DIM = 1024
  const int    col = (int)(j & 1023);
  const float  A   = ws[WS_A4 + row];
  const v4f    x   = *(const v4f*)(x0 + j);
  const v4f    bs  = *(const v4f*)(ws + WS_BSUM + col);
  v4f o;
  o.x = A * x.x + bs.x;
  o.y = A * x.y + bs.y;
  o.z = A * x.z + bs.z;
  o.w = A * x.w + bs.w;
  *(v4f*)(out + j) = o;
}

extern "C" void kernel_launch(void* const* d_in, const int* in_sizes, int n_in,
                              void* d_out, int out_size, void* d_ws, size_t ws_size,
                              hipStream_t stream) {
  const float* x0 = (const float*)d_in[0];   // [16384, 1024] fp32
  const float* W  = (const float*)d_in[1];   // [4, 1024] fp32
  const float* b  = (const float*)d_in[2];   // [4, 1024] fp32
  float* out = (float*)d_out;                // [16384, 1024] fp32
  float* ws  = (float*)d_ws;                 // needs ~1.2 MB

  (void)in_sizes; (void)n_in; (void)out_size; (void)ws_size;

  crossnet_prep<<<1, 1024, 0, stream>>>(W, b, ws);

  // ROWTILES*KCHUNKS = 4096 waves; 8 waves (256 threads) per block
  crossnet_dots<<<(ROWTILES * KCHUNKS) / WPB, 256, 0, stream>>>(x0, W, ws);

  crossnet_rec<<<BATCH / 256, 256, 0, stream>>>(ws);

  crossnet_out<<<(BATCH * DIM / 4) / 256, 256, 0, stream>>>(x0, ws, out);
}